// ChamferLoss_87935160418423
// MI455X (gfx1250) — compile-verified
//
#include <hip/hip_runtime.h>

typedef __attribute__((ext_vector_type(2))) float v2f;
typedef __attribute__((ext_vector_type(8))) float v8f;

#define BATCH 128
#define NP 256
#define MQ 256

__global__ void chamfer_zero_out(float* out) { out[0] = 0.0f; }

__global__ __launch_bounds__(256)
void chamfer_wmma_kernel(const float* __restrict__ p,
                         const float* __restrict__ q,
                         float* __restrict__ out) {
    // Padded 6-vectors (K=0..5 valid, K=6,7 zero); stride 9 floats -> bank-conflict free
    __shared__ float P6[NP][9];
    __shared__ float Q6[MQ][9];
    __shared__ float np2s[NP];
    __shared__ float nq2s[MQ];
    __shared__ float rowMin[NP];
    __shared__ unsigned colMinU[MQ];
    __shared__ float waveSum[8];

    const int b   = blockIdx.x;
    const int tid = threadIdx.x;

    // ---- Stage: build 6-dim vectors (r1,r2,r3,i1,i2,i3) + squared norms ----
    {
        const int n = tid;  // 256 threads cover N == M == 256
        const float4 pr = *(const float4*)(p + ((size_t)(0 * BATCH + b) * NP + n) * 4);
        const float4 pi = *(const float4*)(p + ((size_t)(1 * BATCH + b) * NP + n) * 4);
        P6[n][0] = pr.y; P6[n][1] = pr.z; P6[n][2] = pr.w;
        P6[n][3] = pi.y; P6[n][4] = pi.z; P6[n][5] = pi.w;
        P6[n][6] = 0.0f; P6[n][7] = 0.0f;
        np2s[n] = pr.y*pr.y + pr.z*pr.z + pr.w*pr.w
                + pi.y*pi.y + pi.z*pi.z + pi.w*pi.w;

        const float4 qr = *(const float4*)(q + ((size_t)(0 * BATCH + b) * MQ + n) * 4);
        const float4 qi = *(const float4*)(q + ((size_t)(1 * BATCH + b) * MQ + n) * 4);
        Q6[n][0] = qr.y; Q6[n][1] = qr.z; Q6[n][2] = qr.w;
        Q6[n][3] = qi.y; Q6[n][4] = qi.z; Q6[n][5] = qi.w;
        Q6[n][6] = 0.0f; Q6[n][7] = 0.0f;
        nq2s[n] = qr.y*qr.y + qr.z*qr.z + qr.w*qr.w
                + qi.y*qi.y + qi.z*qi.z + qi.w*qi.w;

        colMinU[n] = 0x7F800000u;  // +inf bits
    }
    __syncthreads();

    const int lane = tid & 31;
    const int wave = tid >> 5;
    const int lcol = lane & 15;
    const int kb   = (lane < 16) ? 0 : 2;  // 32-bit A/B fragment: K pair base per lane half

    // Each wave owns row tiles {wave, wave+8}
    for (int rt = wave; rt < 16; rt += 8) {
        const int row = rt * 16 + lcol;
        v2f a_lo, a_hi;
        a_lo.x = P6[row][kb];     a_lo.y = P6[row][kb + 1];      // K=0..3
        a_hi.x = P6[row][kb + 4]; a_hi.y = P6[row][kb + 5];      // K=4..7 (6,7 are zero)

        const int nbase = rt * 16 + ((lane < 16) ? 0 : 8);
        float np2r[8];
        #pragma unroll
        for (int r = 0; r < 8; ++r) np2r[r] = np2s[nbase + r];

        float rmin[8];
        #pragma unroll
        for (int r = 0; r < 8; ++r) rmin[r] = __builtin_inff();

        for (int ct = 0; ct < 16; ++ct) {
            const int col = ct * 16 + lcol;
            v2f b_lo, b_hi;
            b_lo.x = Q6[col][kb];     b_lo.y = Q6[col][kb + 1];
            b_hi.x = Q6[col][kb + 4]; b_hi.y = Q6[col][kb + 5];
            const float nq2c = nq2s[col];

            v8f c = {};
            c = __builtin_amdgcn_wmma_f32_16x16x4_f32(false, a_lo, false, b_lo,
                                                      (short)0, c, false, false);
            c = __builtin_amdgcn_wmma_f32_16x16x4_f32(false, a_hi, false, b_hi,
                                                      (short)0, c, false, false);

            // dist^2 = |p|^2 + |q|^2 - 2*dot ; clamp tiny negative rounding
            float cmin = __builtin_inff();
            #pragma unroll
            for (int r = 0; r < 8; ++r) {
                float d = fmaxf(fmaf(-2.0f, c[r], np2r[r] + nq2c), 0.0f);
                rmin[r] = fminf(rmin[r], d);
                cmin = fminf(cmin, d);
            }
            // combine the two lane halves (M=0..7 vs M=8..15 share the same column)
            cmin = fminf(cmin, __shfl_xor(cmin, 16, 32));
            if (lane < 16) atomicMin(&colMinU[col], __float_as_uint(cmin));
        }

        // reduce row mins across the 16 lanes of each half
        #pragma unroll
        for (int m = 8; m >= 1; m >>= 1) {
            #pragma unroll
            for (int r = 0; r < 8; ++r)
                rmin[r] = fminf(rmin[r], __shfl_xor(rmin[r], m, 32));
        }
        if (lcol == 0) {
            const int base = rt * 16 + ((lane < 16) ? 0 : 8);
            #pragma unroll
            for (int r = 0; r < 8; ++r) rowMin[base + r] = rmin[r];
        }
    }
    __syncthreads();

    // ---- Sum rowMin + colMin over the batch, then global accumulate ----
    float s = rowMin[tid] + __uint_as_float(colMinU[tid]);
    #pragma unroll
    for (int m = 16; m >= 1; m >>= 1) s += __shfl_xor(s, m, 32);
    if (lane == 0) waveSum[wave] = s;
    __syncthreads();
    if (tid == 0) {
        float t = 0.0f;
        #pragma unroll
        for (int w = 0; w < 8; ++w) t += waveSum[w];
        atomicAdd(out, t);
    }
}

extern "C" void kernel_launch(void* const* d_in, const int* in_sizes, int n_in,
                              void* d_out, int out_size, void* d_ws, size_t ws_size,
                              hipStream_t stream) {
    const float* p = (const float*)d_in[0];
    const float* q = (const float*)d_in[1];
    float* out = (float*)d_out;

    chamfer_zero_out<<<1, 1, 0, stream>>>(out);
    chamfer_wmma_kernel<<<BATCH, 256, 0, stream>>>(p, q, out);
}